// DepthwiseCorrelation_18519898980821
// MI455X (gfx1250) — compile-verified
//
#include <hip/hip_runtime.h>
#include <hip/hip_bf16.h>
#include <stdint.h>

// CDNA5 / gfx1250 depthwise cross-correlation using V_WMMA_F32_16X16X4_F32
// with async global->LDS staging (GLOBAL_LOAD_ASYNC_TO_LDS_B128, ASYNCcnt).
//
// Per (b,c) channel:
//   Phase 1 (WMMA):  R[h, i, wo] = sum_j S[h, wo+j] * T[i, j]
//       M = (wo, h) packed rows, K = j (8, as 2 x K=4), N = i (8 of 16 cols)
//   Phase 2 (VALU):  out[ho, wo] = sum_i R[ho+i, i, wo]
//
// fp32 end-to-end (matches reference precision). Memory-bound: ~224 MB total
// traffic -> ~10 us at 23.3 TB/s; 3.3M wmma instructions sit under that.

typedef __attribute__((ext_vector_type(2))) float v2f;
typedef __attribute__((ext_vector_type(8))) float v8f;

#define NB   64
#define NC   512
#define HT   8
#define WT   8
#define HS   32
#define WS   32
#define HO   25
#define WO   25
#define SLAB (HS * 8 + 4)   // per-wo R slab, padded 4 floats to spread LDS banks

__global__ __launch_bounds__(256, 2) void dwxcorr_wmma_f32(
    const float* __restrict__ tmpl,   // [B, C, 8, 8]
    const float* __restrict__ srch,   // [B, C, 32, 32]
    float* __restrict__ out)          // [B, C, 25, 25]
{
    __shared__ __align__(16) float sS[HS * WS];   // 4 KB search tile
    __shared__ float sT[HT * WT];                 // 256 B template
    __shared__ float sR[WO * SLAB];               // ~25.4 KB phase-1 result

    const int ch   = blockIdx.x;          // b*C + c
    const int tid  = threadIdx.x;
    const int lane = tid & 31;
    const int wave = tid >> 5;

    // ---- Stage S (32x32) straight into LDS with the CDNA5 async path ----
    // 256 lanes x b128 = 4 KB, tracked on ASYNCcnt; no VGPR round trip.
    {
        const float* gp = srch + (size_t)ch * (HS * WS) + tid * 4;
        uint32_t ldsOff = (uint32_t)(uintptr_t)(&sS[tid * 4]);
        asm volatile("global_load_async_to_lds_b128 %0, %1, off"
                     :: "v"(ldsOff), "v"(gp) : "memory");
        if (tid < HT * WT) sT[tid] = tmpl[(size_t)ch * (HT * WT) + tid];
        asm volatile("s_wait_asynccnt 0x0" ::: "memory");
    }
    __syncthreads();

    // ---- Phase 1: 50 M-tiles of 16 rows, 2 x wmma_f32_16x16x4 each ----
    const int r  = lane & 15;   // A: row M within tile ; B: col N (= i) ; D: col N
    const int kh = lane >> 4;   // K half-select for A/B, row-half for D

    // B fragments (template, reused across all 50 tiles): B[k][n] = T[n][k]
    v2f b0, b1;
    {
        const bool nz = (r < 8);                 // only 8 real template rows in N
        const int k0 = 2 * kh;                   // K block 0: j = 0..3
        const int k1 = 4 + 2 * kh;               // K block 1: j = 4..7
        b0.x = nz ? sT[r * WT + k0]     : 0.0f;
        b0.y = nz ? sT[r * WT + k0 + 1] : 0.0f;
        b1.x = nz ? sT[r * WT + k1]     : 0.0f;
        b1.y = nz ? sT[r * WT + k1 + 1] : 0.0f;
    }

    for (int t = wave; t < 50; t += 8) {         // wave-uniform loop (EXEC all 1s)
        const int wo = t >> 1;                   // 0..24
        const int h0 = (t & 1) << 4;             // 0 or 16
        const int rowbase = (h0 + r) * WS + wo;  // A[m][k] = S[h0+m][wo + j]

        v2f a0, a1;
        a0.x = sS[rowbase + 2 * kh];
        a0.y = sS[rowbase + 2 * kh + 1];
        a1.x = sS[rowbase + 4 + 2 * kh];
        a1.y = sS[rowbase + 4 + 2 * kh + 1];

        v8f c = {};
        c = __builtin_amdgcn_wmma_f32_16x16x4_f32(false, a0, false, b0,
                                                  (short)0, c, false, false);
        c = __builtin_amdgcn_wmma_f32_16x16x4_f32(false, a1, false, b1,
                                                  (short)0, c, false, false);

        // D[m][n]: lane col n=r, VGPR v -> row m = v + 8*kh. Keep n<8 (real i).
        if (r < 8) {
            #pragma unroll
            for (int v = 0; v < 8; ++v)
                sR[wo * SLAB + (h0 + v + 8 * kh) * 8 + r] = c[v];
        }
    }
    __syncthreads();

    // ---- Phase 2: diagonal reduction, coalesced non-temporal global store ----
    for (int o = tid; o < HO * WO; o += 256) {
        const int ho = o / WO;
        const int wo = o - ho * WO;
        const float* rp = &sR[wo * SLAB + ho * 8];
        float acc = rp[0];
        #pragma unroll
        for (int i = 1; i < 8; ++i)
            acc += rp[i * 8 + i];
        __builtin_nontemporal_store(acc, &out[(size_t)ch * (HO * WO) + o]);
    }
}

extern "C" void kernel_launch(void* const* d_in, const int* in_sizes, int n_in,
                              void* d_out, int out_size, void* d_ws, size_t ws_size,
                              hipStream_t stream) {
    const float* tmpl = (const float*)d_in[0];   // template_features [64,512,8,8]
    const float* srch = (const float*)d_in[1];   // search_features  [64,512,32,32]
    float* out = (float*)d_out;                  // [64,512,25,25]

    dwxcorr_wmma_f32<<<dim3(NB * NC), dim3(256), 0, stream>>>(tmpl, srch, out);
}